// SupervisedLoss_76338748719202
// MI455X (gfx1250) — compile-verified
//
#include <hip/hip_runtime.h>
#include <hip/hip_bf16.h>

// CDNA5 (gfx1250) fused soft-DTW loss.
// Kernel 1: one workgroup (256 threads = 8 wave32) per cost matrix (1536 total).
//   Phase A: row squared-norms of X and Y (256 rows, one thread each).
//   Phase B: G = X*Y^T via V_WMMA_F32_16X16X4_F32; D = xn + yn - 2G into LDS
//            (row stride 130 floats -> conflict-free anti-diagonal reads).
//   Phase C: soft-DTW DP over anti-diagonals (255 steps), 3 rotating diagonal
//            buffers in LDS, stable softmin3 with gamma = 0.001.
// Kernel 2: single-block reduction -> (E_loss_T0, E_loss0).

typedef __attribute__((ext_vector_type(2))) float v2f;
typedef __attribute__((ext_vector_type(8))) float v8f;

#define GAMMA_ 0.001f
#define INVG_  1000.0f
#define BIG_   1e30f
#define NSEQ   128
#define LDD    130      // padded LDS row stride for the cost tile

__launch_bounds__(256)
__global__ void softdtw_fused_kernel(const float* __restrict__ outputs,
                                     const float* __restrict__ targets,
                                     float* __restrict__ res) {
  __shared__ float Dt[NSEQ * LDD];   // 66,560 B cost tile
  __shared__ float xn[NSEQ];
  __shared__ float yn[NSEQ];
  __shared__ float diag[3 * 132];    // rotating DP diagonals (length 129 used)

  const int mat = blockIdx.x;        // 0..1535, [xy | xx | yy] concatenated
  const int typ = mat >> 9;          // 0: D(x,y), 1: D(x,x), 2: D(y,y)
  const int b   = mat & 511;
  const float* __restrict__ X = (typ == 2 ? targets : outputs) + (size_t)b * NSEQ * NSEQ;
  const float* __restrict__ Y = (typ == 1 ? outputs : targets) + (size_t)b * NSEQ * NSEQ;

  const int t = threadIdx.x;

  // ---------- Phase A: row squared norms ----------
  {
    const float* src = (t < NSEQ) ? (X + t * NSEQ) : (Y + (t - NSEQ) * NSEQ);
    const float4* s4 = (const float4*)src;
    float s = 0.0f;
    #pragma unroll 4
    for (int q = 0; q < NSEQ / 4; ++q) {
      float4 v = s4[q];
      s += v.x * v.x + v.y * v.y + v.z * v.z + v.w * v.w;
    }
    if (t < NSEQ) xn[t] = s; else yn[t - NSEQ] = s;
  }
  __syncthreads();

  // ---------- Phase B: cost tile via f32 WMMA ----------
  // Fragment layout (ISA 7.12.2, 32-bit 16x4 A / 4x16 B): lane<16 holds row
  // (lane), K = {0,1}; lane>=16 holds row (lane-16), K = {2,3}. Identical for
  // A and B, so each fragment is one aligned float2 load per lane.
  {
    const int wave = t >> 5;
    const int lane = t & 31;
    const int half = lane >> 4;
    const int lr   = lane & 15;
    const int tm   = wave * 16;                   // this wave's tile row
    const float* ap = X + (tm + lr) * NSEQ + 2 * half;

    for (int tn = 0; tn < NSEQ; tn += 16) {
      const float* bp = Y + (tn + lr) * NSEQ + 2 * half;
      v8f acc = {0.f, 0.f, 0.f, 0.f, 0.f, 0.f, 0.f, 0.f};
      #pragma unroll 8
      for (int k = 0; k < NSEQ; k += 4) {
        v2f a = *(const v2f*)(ap + k);
        v2f bb = *(const v2f*)(bp + k);
        // (neg_a, A, neg_b, B, c_mod, C, reuse_a, reuse_b)
        acc = __builtin_amdgcn_wmma_f32_16x16x4_f32(false, a, false, bb,
                                                    (short)0, acc, false, false);
      }
      // C layout: lane<16 -> col lane, rows 0..7; lane>=16 -> col lane-16,
      // rows 8..15. The two halves land 16 LDS banks apart (stride 130).
      const int n = tn + lr;
      #pragma unroll
      for (int r = 0; r < 8; ++r) {
        const int m = tm + r + 8 * half;
        Dt[m * LDD + n] = xn[m] + yn[n] - 2.0f * acc[r];
      }
    }
  }
  __syncthreads();

  // ---------- Phase C: soft-DTW DP over anti-diagonals ----------
  // Padded R is (129 x 129): R[0][0]=0, R[0][j>0]=R[i>0][0]=BIG,
  // R[i][j] = D[i-1][j-1] + softmin3(R[i-1][j-1], R[i-1][j], R[i][j-1]).
  // diag k entry j == R[k-j][j].
  float* d0 = diag;          // diag k-2
  float* d1 = diag + 132;    // diag k-1
  float* d2 = diag + 264;    // diag k (current)
  if (t <= NSEQ) {
    d0[t] = (t == 0) ? 0.0f : BIG_;   // diag 0: R[0][0]=0
    d1[t] = BIG_;                     // diag 1: R[1][0]=R[0][1]=BIG
  }
  __syncthreads();

  for (int k = 2; k <= 2 * NSEQ; ++k) {
    if (t <= NSEQ) {
      const int j = t;
      const int i = k - j;
      float v = BIG_;                 // boundary / out-of-range cells
      if (i >= 1 && i <= NSEQ && j >= 1) {
        const float dg = d0[j - 1];   // R[i-1][j-1]
        const float up = d1[j];       // R[i-1][j]
        const float lf = d1[j - 1];   // R[i][j-1]
        const float mn = fminf(dg, fminf(up, lf));
        // stable: m - g*log(sum exp((m-x)/g)); (m-BIG)*1000 -> -1e33 -> exp=0
        const float s = __expf((mn - dg) * INVG_) +
                        __expf((mn - up) * INVG_) +
                        __expf((mn - lf) * INVG_);
        v = Dt[(i - 1) * LDD + (j - 1)] + (mn - GAMMA_ * __logf(s));
      }
      d2[j] = v;
    }
    __syncthreads();
    float* tmp = d0; d0 = d1; d1 = d2; d2 = tmp;  // after: d1 == diag k
  }

  if (t == 0) res[mat] = d1[NSEQ];    // R[128][128]
}

__launch_bounds__(512)
__global__ void softdtw_reduce_kernel(const float* __restrict__ res,
                                      float* __restrict__ out) {
  __shared__ float sm[512];
  const int b = threadIdx.x;
  // per = d_xy - 0.5*(d_xx + d_yy)
  sm[b] = res[b] - 0.5f * (res[512 + b] + res[1024 + b]);
  __syncthreads();
  for (int s = 256; s > 0; s >>= 1) {
    if (b < s) sm[b] += sm[b + s];
    __syncthreads();
  }
  if (b == 0) {
    const float mean = sm[0] * (1.0f / 512.0f);
    out[0] = mean;                          // E_loss_T0
    out[1] = 10.0f * sqrtf(mean + 1e-10f);  // E_loss0
  }
}

extern "C" void kernel_launch(void* const* d_in, const int* in_sizes, int n_in,
                              void* d_out, int out_size, void* d_ws, size_t ws_size,
                              hipStream_t stream) {
  const float* outputs = (const float*)d_in[0];   // (512,128,128) f32
  const float* targets = (const float*)d_in[1];   // (512,128,128) f32
  float* res = (float*)d_ws;                      // 1536 floats of scratch

  softdtw_fused_kernel<<<dim3(1536), dim3(256), 0, stream>>>(outputs, targets, res);
  softdtw_reduce_kernel<<<dim3(1), dim3(512), 0, stream>>>(res, (float*)d_out);
}